// KNNAttentionOptionalLocal_29635274343046
// MI455X (gfx1250) — compile-verified
//
#include <hip/hip_runtime.h>
#include <hip/hip_bf16.h>
#include <math.h>

typedef __attribute__((ext_vector_type(2))) float v2f;
typedef __attribute__((ext_vector_type(8))) float v8f;

#define BATCH  2
#define NQ     2048
#define DIM    512
#define NHEAD  8
#define HDIM   64
#define KNN    32
#define MDB    131072

// ---------------------------------------------------------------------------
// C[M x N] = A[M x K] * W[N x K]^T   (row-major; "NT" GEMM)
// Each wave32 computes a 16(M) x 64(N) strip: one A fragment per k-step feeds
// 4 V_WMMA_F32_16X16X4_F32 against 4 B fragments (distinct accumulators).
// Software pipeline enforced with sched_barrier(0): the 5 loads for step k+4
// are all issued BEFORE the 4 WMMAs of step k, so the WMMAs wait only on the
// previous iteration's (already-landed) loads -> staged s_wait_loadcnt, loads
// for the next step remain in flight underneath the matrix ops.
// A-fragment (16x4 f32): lanes 0-15 hold row M=lane, K=k+0..1;
// lanes 16-31 hold K=k+2..3  -> one aligned float2 per lane.
// C/D layout: VGPR r -> row m0 + r + 8*(lane>=16), col n0 + (lane&15).
// ---------------------------------------------------------------------------
__global__ void gemm_wmma_nt(const float* __restrict__ A,
                             const float* __restrict__ W,
                             float* __restrict__ C,
                             int M, int N, int K) {
    const int wave = threadIdx.x >> 5;
    const int lane = threadIdx.x & 31;
    const int strips_n = N >> 6;                  // 64-wide N strips
    const int strip = blockIdx.x * 4 + wave;      // 4 waves per block
    const int tm  = (strip / strips_n) << 4;
    const int tn0 = (strip % strips_n) << 6;
    if (tm >= M) return;                          // wave-uniform: EXEC stays all-1s

    const int half = lane >> 4;                   // 0: K pair {0,1}, 1: K pair {2,3}
    const int l16  = lane & 15;

    // 32-bit per-lane element offsets from uniform base pointers
    const unsigned ao = (unsigned)(tm + l16) * (unsigned)K + 2u * half;
    const unsigned w0 = (unsigned)(tn0 +  0 + l16) * (unsigned)K + 2u * half;
    const unsigned w1 = (unsigned)(tn0 + 16 + l16) * (unsigned)K + 2u * half;
    const unsigned w2 = (unsigned)(tn0 + 32 + l16) * (unsigned)K + 2u * half;
    const unsigned w3 = (unsigned)(tn0 + 48 + l16) * (unsigned)K + 2u * half;

    v8f acc0 = {}, acc1 = {}, acc2 = {}, acc3 = {};

    // prologue: fragments for k = 0
    v2f a  = *(const v2f*)(A + ao);
    v2f b0 = *(const v2f*)(W + w0);
    v2f b1 = *(const v2f*)(W + w1);
    v2f b2 = *(const v2f*)(W + w2);
    v2f b3 = *(const v2f*)(W + w3);

    for (int k = 0; k < K - 4; k += 4) {
        // prefetch k+4 fragments; keep them ABOVE this step's WMMAs
        v2f an  = *(const v2f*)(A + ao + k + 4);
        v2f b0n = *(const v2f*)(W + w0 + k + 4);
        v2f b1n = *(const v2f*)(W + w1 + k + 4);
        v2f b2n = *(const v2f*)(W + w2 + k + 4);
        v2f b3n = *(const v2f*)(W + w3 + k + 4);

        // fence: scheduler must not sink the loads below / hoist WMMAs above
        __builtin_amdgcn_sched_barrier(0);

        acc0 = __builtin_amdgcn_wmma_f32_16x16x4_f32(false, a, false, b0,
                                                     (short)0, acc0, false, false);
        acc1 = __builtin_amdgcn_wmma_f32_16x16x4_f32(false, a, false, b1,
                                                     (short)0, acc1, false, false);
        acc2 = __builtin_amdgcn_wmma_f32_16x16x4_f32(false, a, false, b2,
                                                     (short)0, acc2, false, false);
        acc3 = __builtin_amdgcn_wmma_f32_16x16x4_f32(false, a, false, b3,
                                                     (short)0, acc3, false, false);
        a = an; b0 = b0n; b1 = b1n; b2 = b2n; b3 = b3n;
    }
    // epilogue: last k-step
    acc0 = __builtin_amdgcn_wmma_f32_16x16x4_f32(false, a, false, b0,
                                                 (short)0, acc0, false, false);
    acc1 = __builtin_amdgcn_wmma_f32_16x16x4_f32(false, a, false, b1,
                                                 (short)0, acc1, false, false);
    acc2 = __builtin_amdgcn_wmma_f32_16x16x4_f32(false, a, false, b2,
                                                 (short)0, acc2, false, false);
    acc3 = __builtin_amdgcn_wmma_f32_16x16x4_f32(false, a, false, b3,
                                                 (short)0, acc3, false, false);

    float* crow = C + (size_t)(tm + 8 * half) * N + tn0 + l16;
#pragma unroll
    for (int r = 0; r < 8; ++r) {
        crow[(size_t)r * N +  0] = acc0[r];
        crow[(size_t)r * N + 16] = acc1[r];
        crow[(size_t)r * N + 32] = acc2[r];
        crow[(size_t)r * N + 48] = acc3[r];
    }
}

// ---------------------------------------------------------------------------
// KNN attention: one wave32 per (b, head, i) query row; lane j <-> key j (K=32).
//  * q row loaded, L2-normalized (fused), staged in LDS (broadcast reads)
//  * per-lane 64-deep dot with gathered key row; prefetch V row meanwhile
//  * softmax over the 32 lanes via shfl_xor butterflies
//  * PV: chunked float4 butterfly reduction across lanes, lane0 stores
// Output written directly in (b, n, h*d) layout (transpose fused).
// ---------------------------------------------------------------------------
__global__ void knn_attn(const float* __restrict__ qbuf,      // (B,NQ,DIM) un-normalized
                         const float* __restrict__ mem_db,    // (B,MDB,2,HDIM)
                         const int* __restrict__ knn_idx,     // (B,NHEAD,NQ,KNN)
                         const unsigned char* __restrict__ mem_mask,
                         const float* __restrict__ scale_param,
                         float* __restrict__ obuf) {           // (B,NQ,DIM)
    __shared__ __align__(16) float s_q[8][HDIM];

    const int wave = threadIdx.x >> 5;
    const int lane = threadIdx.x & 31;
    const int row  = blockIdx.x * 8 + wave;        // over B*NHEAD*NQ = 32768
    const int i    = row % NQ;
    const int bh   = row / NQ;
    const int head = bh % NHEAD;
    const int b    = bh / NHEAD;

    // ---- load + normalize q row (2 floats per lane) ----
    const float* qrow = qbuf + ((size_t)(b * NQ + i) * DIM) + head * HDIM;
    float2 ql = *(const float2*)(qrow + lane * 2);
    float ss = ql.x * ql.x + ql.y * ql.y;
#pragma unroll
    for (int m = 16; m; m >>= 1) ss += __shfl_xor(ss, m, 32);
    const float inv = rsqrtf(fmaxf(ss, 1e-24f));   // 1/clip(||q||,1e-12)
    *(float2*)(&s_q[wave][lane * 2]) = make_float2(ql.x * inv, ql.y * inv);
    __syncthreads();                                // all 8 rows valid, no divergence

    // ---- gather key for this lane, dot with q ----
    const size_t koff = (((size_t)(b * NHEAD + head) * NQ + i) * KNN) + lane;
    const int idx = knn_idx[koff];
    const unsigned char valid = mem_mask[koff];
    const float* kptr = mem_db + (((size_t)b * MDB + idx) * 2 + 0) * HDIM;
    const float* vptr = kptr + HDIM;
    __builtin_prefetch(vptr, 0, 1);                 // global_prefetch_b8 for V row

    float acc = 0.f;
#pragma unroll
    for (int d = 0; d < HDIM; d += 4) {
        float4 kk = *(const float4*)(kptr + d);
        float4 qq = *(const float4*)(&s_q[wave][d]);
        acc = fmaf(kk.x, qq.x, acc);
        acc = fmaf(kk.y, qq.y, acc);
        acc = fmaf(kk.z, qq.z, acc);
        acc = fmaf(kk.w, qq.w, acc);
    }

    const float scale = __expf(scale_param[head]);
    float sim = valid ? acc * scale : -3.402823466e38f;

    // ---- softmax across the 32 keys (one per lane) ----
    float wmax = sim;
#pragma unroll
    for (int m = 16; m; m >>= 1) wmax = fmaxf(wmax, __shfl_xor(wmax, m, 32));
    const float e = __expf(sim - wmax);
    float esum = e;
#pragma unroll
    for (int m = 16; m; m >>= 1) esum += __shfl_xor(esum, m, 32);
    const float attn = e / esum;

    // ---- out[d] = sum_j attn_j * v_j[d], chunked butterfly reduction ----
    float* orow = obuf + ((size_t)(b * NQ + i) * DIM) + head * HDIM;
#pragma unroll
    for (int d = 0; d < HDIM; d += 4) {
        float4 vv = *(const float4*)(vptr + d);
        float4 w = make_float4(vv.x * attn, vv.y * attn, vv.z * attn, vv.w * attn);
#pragma unroll
        for (int m = 16; m; m >>= 1) {
            w.x += __shfl_xor(w.x, m, 32);
            w.y += __shfl_xor(w.y, m, 32);
            w.z += __shfl_xor(w.z, m, 32);
            w.w += __shfl_xor(w.w, m, 32);
        }
        if (lane == 0) *(float4*)(orow + d) = w;
    }
}

// ---------------------------------------------------------------------------
extern "C" void kernel_launch(void* const* d_in, const int* in_sizes, int n_in,
                              void* d_out, int out_size, void* d_ws, size_t ws_size,
                              hipStream_t stream) {
    const float*         x           = (const float*)d_in[0];
    const float*         mem_db      = (const float*)d_in[1];
    const int*           knn_idx     = (const int*)d_in[2];
    const unsigned char* mem_mask    = (const unsigned char*)d_in[3]; // bool, 1 byte
    const float*         Wq          = (const float*)d_in[4];
    // d_in[5] = Wkv: dead code in the reference (kv never used) -- skipped
    const float*         Wout        = (const float*)d_in[6];
    const float*         scale_param = (const float*)d_in[7];
    float*               out         = (float*)d_out;

    float* q_ws = (float*)d_ws;                         // 4096*512 f32 = 8 MB
    float* o_ws = q_ws + (size_t)BATCH * NQ * DIM;      // another 8 MB

    const int Mrows  = BATCH * NQ;                      // 4096
    const int strips = (Mrows / 16) * (DIM / 64);       // 2048 strips, 4 waves/block

    // q = x @ Wq^T   (f32 WMMA, 16x64 strip per wave, pipelined)
    gemm_wmma_nt<<<strips / 4, 128, 0, stream>>>(x, Wq, q_ws, Mrows, DIM, DIM);

    // fused normalize + gather + softmax + PV, output already (b,n,h*d)
    knn_attn<<<(BATCH * NHEAD * NQ) / 8, 256, 0, stream>>>(
        q_ws, mem_db, knn_idx, mem_mask, scale_param, o_ws);

    // final = o @ Wout^T  (f32 WMMA)
    gemm_wmma_nt<<<strips / 4, 128, 0, stream>>>(o_ws, Wout, out, Mrows, DIM, DIM);

    (void)in_sizes; (void)n_in; (void)out_size; (void)ws_size;
}